// DynMoleRouterLoss_15350213116553
// MI455X (gfx1250) — compile-verified
//
#include <hip/hip_runtime.h>
#include <hip/hip_bf16.h>
#include <stdint.h>

// DynMoleRouterLoss on gfx1250 (MI455X), wave32.
//
// ws layout (floats): ws[0]=T (global clip-sum), ws[1]=U (global clip^1.2 sum),
// ws[2..65]   = A[e] = sum_i mask_i * orig_w[i,e]
// ws[66..129] = R[e] = sum_i mask_i * routing_w[i,e]

typedef __attribute__((ext_vector_type(2))) float v2f;
typedef __attribute__((ext_vector_type(8))) float v8f;

#define CHUNK_ROWS 8
#define NEXPERT 64

__device__ __forceinline__ float fast_exp2(float x) { return __builtin_amdgcn_exp2f(x); }

__device__ __forceinline__ float wave_sum(float v) {
#pragma unroll
  for (int o = 16; o; o >>= 1) v += __shfl_xor(v, o, 32);
  return v;
}
__device__ __forceinline__ float wave_max(float v) {
#pragma unroll
  for (int o = 16; o; o >>= 1) v = fmaxf(v, __shfl_xor(v, o, 32));
  return v;
}

struct Acc {
  float T, U, ao0, ao1, ar0, ar1;
};

// One wave processes one row (64 experts, 2 per lane: experts 2*lane, 2*lane+1).
__device__ __forceinline__ void process_row(float a0, float a1, int mv, int lane, Acc& acc) {
  // softmax (max-shifted, like jax.nn.softmax)
  float mx = wave_max(fmaxf(a0, a1));
  float e0 = __expf(a0 - mx);
  float e1 = __expf(a1 - mx);
  float s = wave_sum(e0 + e1);
  float inv = 1.0f / s;
  float w0 = e0 * inv, w1 = e1 * inv;

  // clip + p^1.2 feed both the global entropy accumulators and the row entropy
  float c0 = fmaxf(w0, 1e-5f), c1 = fmaxf(w1, 1e-5f);
  float p0 = fast_exp2(1.2f * __log2f(c0));
  float p1 = fast_exp2(1.2f * __log2f(c1));
  acc.T += c0 + c1;
  acc.U += p0 + p1;

  if (mv != 0) {  // wave-uniform: masked-out rows contribute nothing to the column sums
    float Sr = wave_sum(c0 + c1);
    float Ur = wave_sum(p0 + p1);
    // Tsallis row entropy, q=1.2: (1 - Ur/Sr^q)/(q-1)
    float ent = 5.0f * (1.0f - Ur * fast_exp2(-1.2f * __log2f(Sr)));
    float r0 = w0, r1 = w1;
    if (ent < 1.5f) {  // wave-uniform; rare for near-uniform softmax rows
      // rank(e) = #{w' > w[e]}, sgt(e) = sum of strictly-greater probs,
      // via 32-step lane rotation (covers all 64x64 pairs).
      float b0 = w0, b1 = w1;
      int rk0 = 0, rk1 = 0;
      float sg0 = 0.f, sg1 = 0.f;
      const int src = (lane + 1) & 31;
#pragma unroll 4
      for (int it = 0; it < 32; ++it) {
        if (b0 > w0) { rk0++; sg0 += b0; }
        if (b1 > w0) { rk0++; sg0 += b1; }
        if (b0 > w1) { rk1++; sg1 += b0; }
        if (b1 > w1) { rk1++; sg1 += b1; }
        b0 = __shfl(b0, src, 32);
        b1 = __shfl(b1, src, 32);
      }
      // dropped iff sorted position >= 2 AND cumsum (through itself) > 0.75
      if (rk0 >= 2 && (sg0 + w0) > 0.75f) r0 = 0.f;
      if (rk1 >= 2 && (sg1 + w1) > 0.75f) r1 = 0.f;
    }
    acc.ao0 += w0; acc.ao1 += w1;
    acc.ar0 += r0; acc.ar1 += r1;
  }
}

__global__ __launch_bounds__(256) void moe_loss_init(float* __restrict__ wsf) {
  if (threadIdx.x < 130) wsf[threadIdx.x] = 0.f;
}

__global__ __launch_bounds__(256) void moe_loss_main(const float* __restrict__ logits,
                                                     const int* __restrict__ msk,
                                                     float* __restrict__ wsf,
                                                     int nrows, int ntok) {
  // 8 waves/block, per-wave double-buffered 8-row (2KB) staging buffers: 32KB LDS.
  __shared__ __align__(16) float lbuf[8][2][CHUNK_ROWS * NEXPERT];
  const int lane = threadIdx.x & 31;
  const int wslot = threadIdx.x >> 5;
  const int gwave = (blockIdx.x * blockDim.x + threadIdx.x) >> 5;
  const int nwaves = (gridDim.x * blockDim.x) >> 5;
  const int nchunks = nrows / CHUNK_ROWS;

  Acc acc = {0.f, 0.f, 0.f, 0.f, 0.f, 0.f};

  // CDNA5 async global->LDS stage of one 8-row chunk (2048B = 4 x b128 x 32 lanes)
  auto stage = [&](int chunk, int bufi) {
    const float* gp = logits + (size_t)chunk * (CHUNK_ROWS * NEXPERT) + lane * 4;
    uint32_t ldsb = (uint32_t)(uintptr_t)(&lbuf[wslot][bufi][0]) + (lane << 4);
    asm volatile(
        "global_load_async_to_lds_b128 %0, %1, off\n\t"
        "global_load_async_to_lds_b128 %0, %1, off offset:512\n\t"
        "global_load_async_to_lds_b128 %0, %1, off offset:1024\n\t"
        "global_load_async_to_lds_b128 %0, %1, off offset:1536"
        :: "v"(ldsb), "v"(gp)
        : "memory");
  };

  int cur = gwave;
  int bufi = 0;
  if (cur < nchunks) stage(cur, 0);
  while (cur < nchunks) {
    const int nxt = cur + nwaves;
    if (nxt < nchunks) {
      stage(nxt, bufi ^ 1);
      // 4 newer loads may stay in flight; wait until the 4 for `cur` are done
      asm volatile("s_wait_asynccnt 0x4" ::: "memory");
    } else {
      asm volatile("s_wait_asynccnt 0x0" ::: "memory");
    }
    const float* bp = &lbuf[wslot][bufi][0];
    const int row0 = cur * CHUNK_ROWS;
#pragma unroll
    for (int r = 0; r < CHUNK_ROWS; ++r) {
      const int mv = msk[(row0 + r) % ntok];
      v2f v = *(const v2f*)(bp + r * NEXPERT + lane * 2);
      process_row(v.x, v.y, mv, lane, acc);
    }
    cur = nxt;
    bufi ^= 1;
  }

  // tail rows (nrows not a multiple of CHUNK_ROWS): plain global loads
  for (int row = nchunks * CHUNK_ROWS + gwave; row < nrows; row += nwaves) {
    const int mv = msk[row % ntok];
    v2f v = *(const v2f*)(logits + (size_t)row * NEXPERT + lane * 2);
    process_row(v.x, v.y, mv, lane, acc);
  }

  // commit: wave-reduce scalars, per-lane per-expert column sums
  float T = wave_sum(acc.T);
  float U = wave_sum(acc.U);
  if (lane == 0) {
    atomicAdd(&wsf[0], T);
    atomicAdd(&wsf[1], U);
  }
  atomicAdd(&wsf[2 + 2 * lane + 0], acc.ao0);
  atomicAdd(&wsf[2 + 2 * lane + 1], acc.ao1);
  atomicAdd(&wsf[66 + 2 * lane + 0], acc.ar0);
  atomicAdd(&wsf[66 + 2 * lane + 1], acc.ar1);
}

__global__ __launch_bounds__(256) void moe_loss_finalize(const int* __restrict__ msk,
                                                         int ntok, int layers,
                                                         const float* __restrict__ wsf,
                                                         float* __restrict__ out) {
  __shared__ float red[256];
  __shared__ float sdot;
  const int t = threadIdx.x;

  // denom = layers * sum(mask)
  float s = 0.f;
  for (int i = t; i < ntok; i += 256) s += (float)msk[i];
  red[t] = s;
  __syncthreads();
  for (int st = 128; st; st >>= 1) {
    if (t < st) red[t] += red[t + st];
    __syncthreads();
  }

  // 64-length bilinear form dot(R, A) via V_WMMA_F32_16X16X4_F32 on wave 0.
  // A(16x4): A[m,k] = R[4m+k]; lanes 0-15 hold K=0,1 (v0,v1); lanes 16-31 hold K=2,3.
  // B(4x16): B[k,n] = A[4n+k]; vgpr j: lanes 0-15 -> K=j, lanes 16-31 -> K=j+2.
  // D[m,m] = sum_k R[4m+k]*A[4m+k]; total dot = sum of the 16 diagonal entries.
  if (t < 32) {
    const float* X = wsf + 66;  // R: masked routing_w column sums
    const float* Y = wsf + 2;   // A: masked orig_w column sums
    const int mq = t & 15;
    const int hi = t >> 4;
    v2f a, b;
    a.x = X[4 * mq + 2 * hi + 0];
    a.y = X[4 * mq + 2 * hi + 1];
    b.x = Y[4 * mq + 2 * hi + 0];
    b.y = Y[4 * mq + 2 * hi + 1];
    v8f c = {0.f, 0.f, 0.f, 0.f, 0.f, 0.f, 0.f, 0.f};
    c = __builtin_amdgcn_wmma_f32_16x16x4_f32(false, a, false, b, (short)0, c, false, false);
    // diagonal: D[m,m] for m=0..7 at lane m (vgpr m); m=8..15 at lane m+16, vgpr m-8
    float contrib = 0.f;
    if (t < 8) contrib = c[t];
    else if (t >= 24) contrib = c[t - 24];
    contrib = wave_sum(contrib);
    if (t == 0) sdot = contrib;
  }
  __syncthreads();

  if (t == 0) {
    const float T = wsf[0];
    const float U = wsf[1];
    // global Tsallis entropy, q=1.2: (1 - U/T^1.2)/0.2
    const float ent = 5.0f * (1.0f - U * fast_exp2(-1.2f * __log2f(T)));
    const float denom = (float)layers * red[0];
    const float lb = 64.0f * sdot / (denom * denom);
    out[0] = 1.0e-3f * ent + 1.0e-3f * lb;
  }
}

extern "C" void kernel_launch(void* const* d_in, const int* in_sizes, int n_in,
                              void* d_out, int out_size, void* d_ws, size_t ws_size,
                              hipStream_t stream) {
  (void)n_in; (void)out_size; (void)ws_size;
  const float* logits = (const float*)d_in[0];
  const int* msk = (const int*)d_in[1];
  const int nelems = in_sizes[0];
  const int ntok = in_sizes[1];
  const int nrows = nelems / NEXPERT;
  const int layers = nrows / ntok;
  float* wsf = (float*)d_ws;

  hipLaunchKernelGGL(moe_loss_init, dim3(1), dim3(256), 0, stream, wsf);
  hipLaunchKernelGGL(moe_loss_main, dim3(2048), dim3(256), 0, stream,
                     logits, msk, wsf, nrows, ntok);
  hipLaunchKernelGGL(moe_loss_finalize, dim3(1), dim3(256), 0, stream,
                     msk, ntok, layers, wsf, (float*)d_out);
}